// GraphNN_56968446214400
// MI455X (gfx1250) — compile-verified
//
#include <hip/hip_runtime.h>

typedef __attribute__((ext_vector_type(2))) float v2f;
typedef __attribute__((ext_vector_type(8))) float v8f;

#define FDIM 64   // aggregation feature width (both layers aggregate 64-dim data)

// deg[i] = 1 (self-loop contribution)
__global__ void k_init_deg(float* __restrict__ deg, int n) {
    int i = blockIdx.x * blockDim.x + threadIdx.x;
    if (i < n) deg[i] = 1.0f;
}

// deg[dst[e]] += 1
__global__ void k_deg_accum(const int* __restrict__ dst, float* __restrict__ deg, int E) {
    int e = blockIdx.x * blockDim.x + threadIdx.x;
    if (e < E) atomicAdd(&deg[dst[e]], 1.0f);
}

// dinv[i] = deg>0 ? rsqrt(deg) : 0
__global__ void k_dinv(const float* __restrict__ deg, float* __restrict__ dinv, int n) {
    int i = blockIdx.x * blockDim.x + threadIdx.x;
    if (i < n) { float d = deg[i]; dinv[i] = d > 0.0f ? rsqrtf(d) : 0.0f; }
}

// agg[i][f] = x[i][f] * dinv[i]^2  (+ bias[f] if given)  -- also serves as initializer
__global__ void k_selfloop(const float* __restrict__ x, const float* __restrict__ dinv,
                           const float* __restrict__ bias, float* __restrict__ agg, int n) {
    int idx = blockIdx.x * blockDim.x + threadIdx.x;
    if (idx < n * FDIM) {
        int i = idx >> 6;
        int f = idx & 63;
        float di = dinv[i];
        float v  = x[idx] * di * di;
        if (bias) v += bias[f];
        agg[idx] = v;
    }
}

// One wave32 per edge; lane handles features 2*lane, 2*lane+1 (float2 coalesced).
__global__ void k_scatter(const float* __restrict__ x, const int* __restrict__ src,
                          const int* __restrict__ dst, const float* __restrict__ dinv,
                          float* __restrict__ agg, int E) {
    long long gtid = (long long)blockIdx.x * blockDim.x + threadIdx.x;
    int e    = (int)(gtid >> 5);
    int lane = (int)(gtid & 31);
    if (e < E) {
        int s = src[e], d = dst[e];
        float w = dinv[s] * dinv[d];
        v2f xv = *(const v2f*)(x + (long long)s * FDIM + lane * 2);
        float* p = agg + (long long)d * FDIM + lane * 2;
        atomicAdd(p,     xv.x * w);
        atomicAdd(p + 1, xv.y * w);
    }
}

// out[n x ncols] = act(A[n x K] @ Bm[K x ncols] + bias), fp32 WMMA 16x16x4.
// blockDim = (32, 8): one wave per 16x16 output tile; tile guard is wave-uniform.
__global__ void k_gemm_wmma(const float* __restrict__ A, const float* __restrict__ Bm,
                            const float* __restrict__ bias, float* __restrict__ out,
                            int n, int K, int ncols, int relu) {
    int col_tile = blockIdx.x;
    int row_tile = blockIdx.y * blockDim.y + threadIdx.y;
    if (row_tile * 16 >= n) return;          // wave-uniform: EXEC stays all-1s

    int lane = threadIdx.x;                  // 0..31
    int half = lane >> 4;                    // 0: K+0/K+1, 1: K+2/K+3
    int m    = lane & 15;
    int row0 = row_tile * 16;
    int col0 = col_tile * 16;

    const float* arow = A  + (long long)(row0 + m) * K + 2 * half;   // A[row0+m][2*half ..]
    const float* bcol = Bm + (long long)(2 * half) * ncols + col0 + m;

    v8f acc = {};
    for (int k0 = 0; k0 < K; k0 += 4) {
        // A 16x4: lane<16 holds K=k0,k0+1 of row m; lane>=16 holds K=k0+2,k0+3
        v2f a = *(const v2f*)(arow + k0);
        // B 4x16: lane<16 holds rows k0,k0+1 at col m; lane>=16 holds rows k0+2,k0+3
        v2f b;
        b.x = bcol[(long long)(k0)     * ncols];
        b.y = bcol[(long long)(k0 + 1) * ncols];
        acc = __builtin_amdgcn_wmma_f32_16x16x4_f32(
                  false, a, false, b, (short)0, acc, false, false);
    }

    float bv = bias ? bias[col0 + m] : 0.0f;
#pragma unroll
    for (int r = 0; r < 8; ++r) {
        // D layout: VGPR r -> row r (lanes 0-15) / row r+8 (lanes 16-31), col = m
        float v = acc[r] + bv;
        if (relu) v = fmaxf(v, 0.0f);
        out[(long long)(row0 + r + 8 * half) * ncols + col0 + m] = v;
    }
}

extern "C" void kernel_launch(void* const* d_in, const int* in_sizes, int n_in,
                              void* d_out, int out_size, void* d_ws, size_t ws_size,
                              hipStream_t stream) {
    (void)n_in; (void)out_size; (void)ws_size;

    const int*   eidx = (const int*)d_in[0];    // [2, E] int32
    const float* emb  = (const float*)d_in[1];  // [n, 64]
    const float* W1   = (const float*)d_in[2];  // [64, 128]
    const float* b1   = (const float*)d_in[3];  // [128]
    const float* W2   = (const float*)d_in[4];  // [128, 64]
    const float* b2   = (const float*)d_in[5];  // [64]

    const int E = in_sizes[0] / 2;
    const int n = in_sizes[1] / 64;             // 100000
    const int* src = eidx;
    const int* dst = eidx + E;

    // Workspace layout (floats): deg[n] | dinv[n] | agg1/t[n*64] | h[n*128]
    float* deg  = (float*)d_ws;
    float* dinv = deg + n;
    float* agg1 = dinv + n;                     // reused for t after GEMM1 consumes it
    float* h    = agg1 + (size_t)n * 64;
    float* t    = agg1;
    float* out  = (float*)d_out;

    const int B = 256;
    // --- degrees & normalization ---
    k_init_deg <<<(n + B - 1) / B, B, 0, stream>>>(deg, n);
    k_deg_accum<<<(E + B - 1) / B, B, 0, stream>>>(dst, deg, E);
    k_dinv     <<<(n + B - 1) / B, B, 0, stream>>>(deg, dinv, n);

    // --- layer 1: agg1 = A_norm @ emb, then h = relu(agg1 @ W1 + b1) ---
    k_selfloop<<<(int)(((size_t)n * 64 + B - 1) / B), B, 0, stream>>>(emb, dinv, nullptr, agg1, n);
    k_scatter <<<(int)(((size_t)E * 32 + B - 1) / B), B, 0, stream>>>(emb, src, dst, dinv, agg1, E);
    {
        dim3 block(32, 8);
        dim3 grid(128 / 16, ((n + 15) / 16 + 7) / 8);
        k_gemm_wmma<<<grid, block, 0, stream>>>(agg1, W1, b1, h, n, 64, 128, 1);
    }

    // --- layer 2: t = h @ W2, then out = A_norm @ t + b2 ---
    {
        dim3 block(32, 8);
        dim3 grid(64 / 16, ((n + 15) / 16 + 7) / 8);
        k_gemm_wmma<<<grid, block, 0, stream>>>(h, W2, nullptr, t, n, 128, 64, 0);
    }
    k_selfloop<<<(int)(((size_t)n * 64 + B - 1) / B), B, 0, stream>>>(t, dinv, b2, out, n);
    k_scatter <<<(int)(((size_t)E * 32 + B - 1) / B), B, 0, stream>>>(t, src, dst, dinv, out, E);
}